// GAT_block_23819888623726
// MI455X (gfx1250) — compile-verified
//
#include <hip/hip_runtime.h>
#include <hip/hip_bf16.h>

// Problem constants (from reference)
#define N_NODES 50000
#define N_EDGES 1600000
#define F_DIM   128
#define H_HEADS 8
#define D_HEAD  16
#define E_DIM   16
#define HD      (H_HEADS * D_HEAD)       // 128
#define ET      (N_EDGES + N_NODES)      // edges incl. self loops
#define EPS_IN  1e-5f
#define NEG_SLOPE 0.2f

typedef float v2f __attribute__((ext_vector_type(2)));
typedef float v8f __attribute__((ext_vector_type(8)));

// ---------------------------------------------------------------------------
// K0: initialize accumulators in workspace
// ---------------------------------------------------------------------------
__global__ void k_init(float* deg, float* loopattr, float* denom, float* outacc,
                       int* amaxK) {
    int stride = gridDim.x * blockDim.x;
    int tid = blockIdx.x * blockDim.x + threadIdx.x;
    for (int i = tid; i < N_NODES; i += stride) deg[i] = 0.0f;
    for (int i = tid; i < N_NODES * E_DIM; i += stride) loopattr[i] = 0.0f;
    for (int i = tid; i < N_NODES * H_HEADS; i += stride) denom[i] = 0.0f;
    for (int i = tid; i < N_NODES * HD; i += stride) outacc[i] = 0.0f;
    for (int i = tid; i < N_NODES * H_HEADS; i += stride) amaxK[i] = (int)0x80000000;
}

// ---------------------------------------------------------------------------
// K1: degree + per-node sum of incoming edge_attr (for fill_value='mean')
// ---------------------------------------------------------------------------
__global__ void k_deg_loop(const int* __restrict__ dst,
                           const float* __restrict__ edge_attr,
                           float* deg, float* loopsum) {
    int e = blockIdx.x * blockDim.x + threadIdx.x;
    if (e >= N_EDGES) return;
    int t = dst[e];
    atomicAdd(&deg[t], 1.0f);
    const float* ea = edge_attr + (size_t)e * E_DIM;
    float* ls = loopsum + (size_t)t * E_DIM;
#pragma unroll
    for (int k = 0; k < E_DIM; ++k) atomicAdd(&ls[k], ea[k]);
}

__global__ void k_loop_final(float* loopattr, const float* __restrict__ deg) {
    int idx = blockIdx.x * blockDim.x + threadIdx.x;
    if (idx >= N_NODES * E_DIM) return;
    float d = deg[idx >> 4];
    d = d < 1.0f ? 1.0f : d;
    loopattr[idx] /= d;
}

// ---------------------------------------------------------------------------
// K2: h = x @ W  via V_WMMA_F32_16X16X4_F32 (wave32, 16x16 tile per wave)
//   A (16x4 f32): lane m = lane&15, VGPR0/1 hold K = (lane>>4)*2 + {0,1}
//   B (4x16 f32): lane n = lane&15, VGPR0/1 hold K = (lane>>4)*2 + {0,1}
//   C (16x16 f32): VGPR r holds M = r + (lane>>4)*8, N = lane&15
// ---------------------------------------------------------------------------
__global__ void __launch_bounds__(256)
k_gemm_h(const float* __restrict__ x, const float* __restrict__ W,
         float* __restrict__ h) {
    int wave = threadIdx.x >> 5;
    int lane = threadIdx.x & 31;
    int tile = blockIdx.x * 8 + wave;         // 25000 tiles total
    int tileM = tile >> 3;                    // 3125 row tiles
    int tileN = tile & 7;                     // 8 col tiles
    int m   = tileM * 16 + (lane & 15);
    int kb  = (lane >> 4) << 1;               // 0 or 2
    int col = tileN * 16 + (lane & 15);

    v8f acc = {};
#pragma unroll
    for (int k = 0; k < F_DIM; k += 4) {
        v2f a, b;
        a.x = x[(size_t)m * F_DIM + k + kb];
        a.y = x[(size_t)m * F_DIM + k + kb + 1];
        b.x = W[(size_t)(k + kb) * HD + col];
        b.y = W[(size_t)(k + kb + 1) * HD + col];
        acc = __builtin_amdgcn_wmma_f32_16x16x4_f32(
            /*neg_a=*/false, a, /*neg_b=*/false, b,
            /*c_mod=*/(short)0, acc, /*reuse_a=*/false, /*reuse_b=*/false);
    }
    int rbase = tileM * 16 + ((lane >> 4) << 3);
#pragma unroll
    for (int r = 0; r < 8; ++r)
        h[(size_t)(rbase + r) * HD + col] = acc[r];
}

// ---------------------------------------------------------------------------
// K3: per-node attention logit contributions a_src, a_dst
// ---------------------------------------------------------------------------
__global__ void k_attn_node(const float* __restrict__ h,
                            const float* __restrict__ att_src,
                            const float* __restrict__ att_dst,
                            float* a_src, float* a_dst) {
    int idx = blockIdx.x * blockDim.x + threadIdx.x;   // n*H + hh
    if (idx >= N_NODES * H_HEADS) return;
    int n = idx >> 3, hh = idx & 7;
    const float* hp = h + (size_t)n * HD + hh * D_HEAD;
    const float* as = att_src + hh * D_HEAD;
    const float* ad = att_dst + hh * D_HEAD;
    float s = 0.f, d = 0.f;
#pragma unroll
    for (int i = 0; i < D_HEAD; ++i) {
        float v = hp[i];
        s += v * as[i];
        d += v * ad[i];
    }
    a_src[idx] = s;
    a_dst[idx] = d;
}

// ---------------------------------------------------------------------------
// K3b: fold W_e and att_edge into a 16x8 matrix:  a_edge = ea @ V
// ---------------------------------------------------------------------------
__global__ void k_edge_proj(const float* __restrict__ W_e,
                            const float* __restrict__ att_edge, float* V) {
    int idx = threadIdx.x;          // 128 threads: (k, hh)
    int k = idx >> 3, hh = idx & 7;
    float s = 0.f;
#pragma unroll
    for (int d = 0; d < D_HEAD; ++d)
        s += W_e[(size_t)k * HD + hh * D_HEAD + d] * att_edge[hh * D_HEAD + d];
    V[idx] = s;
}

// ---------------------------------------------------------------------------
// K4: edge logits (leaky_relu) + segment max via monotonic-int atomicMax
// ---------------------------------------------------------------------------
__global__ void k_logits(const int* __restrict__ src, const int* __restrict__ dst,
                         const float* __restrict__ edge_attr,
                         const float* __restrict__ loopattr,
                         const float* __restrict__ Vc,
                         const float* __restrict__ a_src,
                         const float* __restrict__ a_dst,
                         float* alpha, int* amaxK) {
    int i = blockIdx.x * blockDim.x + threadIdx.x;
    if (i >= ET) return;
    int s, t;
    const float* ea;
    if (i < N_EDGES) {
        s = src[i]; t = dst[i];
        ea = edge_attr + (size_t)i * E_DIM;
    } else {
        s = i - N_EDGES; t = s;
        ea = loopattr + (size_t)s * E_DIM;
    }
    float eav[E_DIM];
#pragma unroll
    for (int k = 0; k < E_DIM; ++k) eav[k] = ea[k];
#pragma unroll
    for (int hh = 0; hh < H_HEADS; ++hh) {
        float ae = 0.f;
#pragma unroll
        for (int k = 0; k < E_DIM; ++k) ae += eav[k] * Vc[k * 8 + hh];
        float l = a_src[s * 8 + hh] + a_dst[t * 8 + hh] + ae;
        l = l > 0.f ? l : NEG_SLOPE * l;
        alpha[(size_t)i * 8 + hh] = l;
        int bi = __float_as_int(l);
        int key = bi >= 0 ? bi : (bi ^ 0x7FFFFFFF);
        atomicMax(&amaxK[t * 8 + hh], key);
    }
}

// ---------------------------------------------------------------------------
// K5: alpha = exp(logit - max); denom = segment_sum(alpha)
// ---------------------------------------------------------------------------
__global__ void k_expsum(const int* __restrict__ dst, float* alpha,
                         const int* __restrict__ amaxK, float* denom) {
    int i = blockIdx.x * blockDim.x + threadIdx.x;
    if (i >= ET) return;
    int t = (i < N_EDGES) ? dst[i] : (i - N_EDGES);
#pragma unroll
    for (int hh = 0; hh < H_HEADS; ++hh) {
        int key = amaxK[t * 8 + hh];
        int bi = key >= 0 ? key : (key ^ 0x7FFFFFFF);
        float mx = __int_as_float(bi);
        float a = __expf(alpha[(size_t)i * 8 + hh] - mx);
        alpha[(size_t)i * 8 + hh] = a;
        atomicAdd(&denom[t * 8 + hh], a);
    }
}

// ---------------------------------------------------------------------------
// K6: weighted scatter-add of source features. 4 threads/edge, 32 ch each.
// ---------------------------------------------------------------------------
__global__ void k_scatter(const int* __restrict__ src, const int* __restrict__ dst,
                          const float* __restrict__ alpha,
                          const float* __restrict__ denom,
                          const float* __restrict__ h, float* outacc) {
    long long tid = (long long)blockIdx.x * blockDim.x + threadIdx.x;
    if (tid >= (long long)ET * 4) return;
    int g = (int)(tid & 3);
    long long i = tid >> 2;
    int s, t;
    if (i < N_EDGES) { s = src[i]; t = dst[i]; }
    else             { s = (int)(i - N_EDGES); t = s; }
#pragma unroll
    for (int hh = g * 2; hh < g * 2 + 2; ++hh) {
        float w = alpha[(size_t)i * 8 + hh] / (denom[t * 8 + hh] + 1e-16f);
        const float* hp = h + (size_t)s * HD + hh * D_HEAD;
        float* op = outacc + (size_t)t * HD + hh * D_HEAD;
#pragma unroll
        for (int d = 0; d < D_HEAD; ++d) atomicAdd(&op[d], hp[d] * w);
    }
}

// ---------------------------------------------------------------------------
// K7: InstanceNorm stats per channel (LDS tree reduction -> ds ops)
// ---------------------------------------------------------------------------
__global__ void k_stats(const float* __restrict__ outacc, float* meanb,
                        float* invstd) {
    __shared__ float ssum[256];
    __shared__ float ssq[256];
    int c = blockIdx.x;               // channel 0..127
    float s = 0.f, q = 0.f;
    for (int n = threadIdx.x; n < N_NODES; n += blockDim.x) {
        float v = outacc[(size_t)n * HD + c];
        s += v;
        q += v * v;
    }
    ssum[threadIdx.x] = s;
    ssq[threadIdx.x] = q;
    __syncthreads();
    for (int off = 128; off > 0; off >>= 1) {
        if (threadIdx.x < off) {
            ssum[threadIdx.x] += ssum[threadIdx.x + off];
            ssq[threadIdx.x]  += ssq[threadIdx.x + off];
        }
        __syncthreads();
    }
    if (threadIdx.x == 0) {
        float m = ssum[0] / (float)N_NODES;
        float var = ssq[0] / (float)N_NODES - m * m;
        meanb[c] = m;
        invstd[c] = rsqrtf(var + EPS_IN);
    }
}

// ---------------------------------------------------------------------------
// K8: normalize + affine + residual + ELU
// ---------------------------------------------------------------------------
__global__ void k_final(const float* __restrict__ x,
                        const float* __restrict__ outacc,
                        const float* __restrict__ meanb,
                        const float* __restrict__ invstd,
                        const float* __restrict__ gamma,
                        const float* __restrict__ beta, float* out) {
    int idx = blockIdx.x * blockDim.x + threadIdx.x;
    if (idx >= N_NODES * HD) return;
    int c = idx & 127;
    float normed = (outacc[idx] - meanb[c]) * invstd[c] * gamma[c] + beta[c];
    float t = x[idx] + normed;
    out[idx] = t > 0.f ? t : expm1f(t);
}

// ---------------------------------------------------------------------------
extern "C" void kernel_launch(void* const* d_in, const int* in_sizes, int n_in,
                              void* d_out, int out_size, void* d_ws, size_t ws_size,
                              hipStream_t stream) {
    const float* x        = (const float*)d_in[0];
    const int*   ei       = (const int*)d_in[1];   // (2, E) row-major
    const float* edgeattr = (const float*)d_in[2];
    const float* W        = (const float*)d_in[3];
    const float* att_src  = (const float*)d_in[4];
    const float* att_dst  = (const float*)d_in[5];
    const float* W_e      = (const float*)d_in[6];
    const float* att_edge = (const float*)d_in[7];
    const float* gamma    = (const float*)d_in[8];
    const float* beta     = (const float*)d_in[9];
    float* out = (float*)d_out;

    const int* src = ei;
    const int* dst = ei + N_EDGES;

    // Workspace carve-up (floats)
    float* ws = (float*)d_ws;
    float* hbuf     = ws;                               // N*HD      = 6.4M
    float* a_src    = hbuf   + (size_t)N_NODES * HD;    // N*H
    float* a_dst    = a_src  + (size_t)N_NODES * H_HEADS;
    float* degb     = a_dst  + (size_t)N_NODES * H_HEADS;
    float* loopattr = degb   + N_NODES;                 // N*ED
    float* Vc       = loopattr + (size_t)N_NODES * E_DIM; // 128
    float* alpha    = Vc + 128;                         // ET*H = 13.2M
    float* denom    = alpha + (size_t)ET * H_HEADS;     // N*H
    float* outacc   = denom + (size_t)N_NODES * H_HEADS;// N*HD
    float* meanb    = outacc + (size_t)N_NODES * HD;    // 128
    float* invstd   = meanb + 128;                      // 128
    int*   amaxK    = (int*)(invstd + 128);             // N*H

    // K0: init accumulators
    k_init<<<1024, 256, 0, stream>>>(degb, loopattr, denom, outacc, amaxK);

    // K1: degree + loop-attr sums
    k_deg_loop<<<(N_EDGES + 255) / 256, 256, 0, stream>>>(dst, edgeattr, degb, loopattr);
    k_loop_final<<<(N_NODES * E_DIM + 255) / 256, 256, 0, stream>>>(loopattr, degb);

    // K2: h = x @ W via FP32 WMMA (3125 blocks x 8 waves = 25000 16x16 tiles)
    k_gemm_h<<<(N_NODES / 16) * 8 / 8, 256, 0, stream>>>(x, W, hbuf);

    // K3: node attention scalars; K3b: folded edge projection
    k_attn_node<<<(N_NODES * H_HEADS + 255) / 256, 256, 0, stream>>>(
        hbuf, att_src, att_dst, a_src, a_dst);
    k_edge_proj<<<1, 128, 0, stream>>>(W_e, att_edge, Vc);

    // K4..K6: edge softmax + aggregation
    k_logits<<<(ET + 255) / 256, 256, 0, stream>>>(src, dst, edgeattr, loopattr,
                                                   Vc, a_src, a_dst, alpha, amaxK);
    k_expsum<<<(ET + 255) / 256, 256, 0, stream>>>(dst, alpha, amaxK, denom);
    {
        long long tot = (long long)ET * 4;
        int blocks = (int)((tot + 255) / 256);
        k_scatter<<<blocks, 256, 0, stream>>>(src, dst, alpha, denom, hbuf, outacc);
    }

    // K7: InstanceNorm stats; K8: finalize
    k_stats<<<HD, 256, 0, stream>>>(outacc, meanb, invstd);
    k_final<<<(N_NODES * HD + 255) / 256, 256, 0, stream>>>(
        x, outacc, meanb, invstd, gamma, beta, out);
}